// LSS_Core_87419764343041
// MI455X (gfx1250) — compile-verified
//
#include <hip/hip_runtime.h>

typedef __attribute__((ext_vector_type(2))) float v2f;
typedef __attribute__((ext_vector_type(8))) float v8f;

#define B_    4
#define N_    6
#define CIN   512
#define H_    16
#define W_    44
#define HW    704        // H_*W_
#define DD    41         // depth bins
#define CC    64         // context channels
#define KOUT  105        // DD + CC
#define KPAD  112        // 7 WMMA tiles of 16
#define GRID_ 200
#define GEO_FLOATS 512   // ws offset (floats) where packed W begins

// ---------------------------------------------------------------------------
// Kernel 0: per-camera geometry — M3 = rots @ inv(K), plus trans (12 floats).
// ---------------------------------------------------------------------------
__global__ void lss_prep(const float* __restrict__ rots,
                         const float* __restrict__ trans,
                         const float* __restrict__ intr,
                         float* __restrict__ geo) {
    int cam = threadIdx.x;
    if (cam >= B_ * N_) return;
    const float* K = intr + cam * 9;
    float a = K[0], b = K[1], c = K[2];
    float d = K[3], e = K[4], f = K[5];
    float g = K[6], h = K[7], i = K[8];
    float det = a * (e * i - f * h) - b * (d * i - f * g) + c * (d * h - e * g);
    float id = 1.0f / det;
    float inv[9];
    inv[0] =  (e * i - f * h) * id;
    inv[1] = -(b * i - c * h) * id;
    inv[2] =  (b * f - c * e) * id;
    inv[3] = -(d * i - f * g) * id;
    inv[4] =  (a * i - c * g) * id;
    inv[5] = -(a * f - c * d) * id;
    inv[6] =  (d * h - e * g) * id;
    inv[7] = -(a * h - b * g) * id;
    inv[8] =  (a * e - b * d) * id;
    const float* R = rots + cam * 9;
    float* o = geo + cam * 12;
    for (int r = 0; r < 3; ++r)
        for (int cj = 0; cj < 3; ++cj)
            o[r * 3 + cj] = R[r * 3 + 0] * inv[0 * 3 + cj] +
                            R[r * 3 + 1] * inv[1 * 3 + cj] +
                            R[r * 3 + 2] * inv[2 * 3 + cj];
    o[9]  = trans[cam * 3 + 0];
    o[10] = trans[cam * 3 + 1];
    o[11] = trans[cam * 3 + 2];
}

// ---------------------------------------------------------------------------
// Kernel P: pack W_enc (512 x 105) into K-pair-interleaved, zero-padded
// Wpad[(k>>1)*KPAD + n] = { W[k][n], W[k+1][n] }  (n >= 105 -> 0).
// Makes every B fragment a single unguarded global_load_b64 in the hot loop.
// ---------------------------------------------------------------------------
__global__ void lss_pack(const float* __restrict__ Wenc, v2f* __restrict__ Wpad) {
    const int idx = blockIdx.x * 256 + threadIdx.x;       // 0 .. 256*112-1
    if (idx >= (CIN / 2) * KPAD) return;
    const int kp = idx / KPAD;                            // k pair
    const int n  = idx - kp * KPAD;
    v2f v;
    if (n < KOUT) {
        v.x = Wenc[(size_t)(2 * kp)     * KOUT + n];
        v.y = Wenc[(size_t)(2 * kp + 1) * KOUT + n];
    } else {
        v.x = 0.0f; v.y = 0.0f;
    }
    Wpad[idx] = v;
}

// ---------------------------------------------------------------------------
// Kernel Z: zero the BEV output (harness poisons d_out).
// ---------------------------------------------------------------------------
__global__ void lss_zero(float4* __restrict__ out) {
    out[(size_t)blockIdx.x * blockDim.x + threadIdx.x] = make_float4(0.f, 0.f, 0.f, 0.f);
}

// ---------------------------------------------------------------------------
// Main fused kernel: GEMM (fp32 WMMA) + softmax + outer-product + geometry +
// scatter-add. One block per 16-pixel tile of one camera. 256 thr = 8 waves.
// ---------------------------------------------------------------------------
__global__ __launch_bounds__(256) void lss_main(const float* __restrict__ x,
                                                const v2f*  __restrict__ Wpad,
                                                const float* __restrict__ benc,
                                                const float* __restrict__ geo,
                                                float* __restrict__ out) {
    __shared__ float feats[16 * KPAD];   // 16 pixels x 112 features
    __shared__ int   slin[16 * DD];      // packed (pixel<<16 | lin), -1 invalid
    __shared__ float swgt[16 * DD];      // depth prob per point

    const int tid  = threadIdx.x;
    const int lane = tid & 31;
    const int wv   = tid >> 5;                 // wave 0..7
    const int cam  = blockIdx.x / (HW / 16);   // 0..23
    const int pix0 = (blockIdx.x % (HW / 16)) * 16;

    // ---- seed LDS feats with bias (0 in padded lanes) ----
    for (int i = tid; i < 16 * KPAD; i += 256) {
        int kq = i % KPAD;
        feats[i] = (kq < KOUT) ? benc[kq] : 0.0f;
    }
    __syncthreads();

    // ---- GEMM: D[16x112] += A[16x512] * W[512x112], K split over 8 waves.
    // A fragment (16x4 fp32): lanes 0-15 hold M, K={0,1}; lanes 16-31 K={2,3}.
    // B fragment (4x16 fp32): lane carries N=lane%16, same K striping; the
    // packed layout delivers {W[k][n], W[k+1][n]} in one b64 load.
    const int m     = lane & 15;
    const int kh    = lane >> 4;               // 0 or 1 (K half)
    const float* xb = x + (size_t)cam * CIN * HW + pix0 + m;
    const v2f* wb   = Wpad + m;                // + kpair*KPAD + tile*16

    v8f acc[7];
#pragma unroll
    for (int t = 0; t < 7; ++t) acc[t] = (v8f){0.f, 0.f, 0.f, 0.f, 0.f, 0.f, 0.f, 0.f};

    for (int kk = 0; kk < 16; ++kk) {
        const int k  = wv * 64 + kk * 4 + kh * 2;   // even
        const int kp = k >> 1;                       // k pair index
        v2f a;
        a.x = xb[(size_t)k * HW];
        a.y = xb[(size_t)(k + 1) * HW];
        if (kk < 15) __builtin_prefetch(xb + (size_t)(k + 4) * HW, 0, 3);
#pragma unroll
        for (int t = 0; t < 7; ++t) {
            const v2f bf = wb[(size_t)kp * KPAD + t * 16];
            acc[t] = __builtin_amdgcn_wmma_f32_16x16x4_f32(
                false, a, false, bf, (short)0, acc[t], false, false);
        }
    }

    // ---- reduce the 8 wave partials into LDS (ds_add_f32) ----
#pragma unroll
    for (int t = 0; t < 7; ++t) {
#pragma unroll
        for (int r = 0; r < 8; ++r) {
            const int M = r + 8 * kh;            // C/D layout: VGPR r -> row
            atomicAdd(&feats[M * KPAD + t * 16 + m], acc[t][r]);
        }
    }
    __syncthreads();

    // ---- softmax over the 41 depth logits; one pixel per 16-lane group ----
    {
        const int p = wv * 2 + kh;
        const int l = m;
        float* row = &feats[p * KPAD];
        float v0 = row[l];
        float v1 = row[l + 16];
        const bool has2 = (l + 32) < DD;
        float v2 = has2 ? row[l + 32] : -3.402823466e38f;
        float mx = fmaxf(v0, fmaxf(v1, v2));
#pragma unroll
        for (int o = 8; o; o >>= 1) mx = fmaxf(mx, __shfl_xor(mx, o, 16));
        float e0 = expf(v0 - mx);
        float e1 = expf(v1 - mx);
        float e2 = has2 ? expf(v2 - mx) : 0.0f;
        float s = e0 + e1 + e2;
#pragma unroll
        for (int o = 8; o; o >>= 1) s += __shfl_xor(s, o, 16);
        const float inv = 1.0f / s;
        row[l]      = e0 * inv;
        row[l + 16] = e1 * inv;
        if (has2) row[l + 32] = e2 * inv;
    }
    __syncthreads();

    // ---- geometry: (pixel, d) -> BEV cell + weight ----
    const float* g = geo + cam * 12;
    const float g0 = g[0], g1 = g[1], g2 = g[2];
    const float g3 = g[3], g4 = g[4], g5 = g[5];
    const float g6 = g[6], g7 = g[7], g8 = g[8];
    const float tx = g[9], ty = g[10], tz = g[11];

    for (int pt = tid; pt < 16 * DD; pt += 256) {
        const int pixel = pt / DD;
        const int d     = pt - pixel * DD;
        const int pg    = pix0 + pixel;
        const float u   = (float)(pg % W_);
        const float v   = (float)(pg / W_);
        const float dep = 4.0f + (float)d;
        const float ex = (g0 * u + g1 * v + g2) * dep + tx;
        const float ey = (g3 * u + g4 * v + g5) * dep + ty;
        const float ez = (g6 * u + g7 * v + g8) * dep + tz;
        // trunc-toward-zero matches astype(int32)
        const int xi = (int)((ex + 50.0f) * 2.0f);
        const int yi = (int)((ey + 50.0f) * 2.0f);
        const int zi = (int)((ez + 50.0f) * 2.0f);
        const bool valid = (xi >= 0) & (xi < GRID_) &
                           (yi >= 0) & (yi < GRID_) & (zi == 0);
        slin[pt] = valid ? ((pixel << 16) | (yi * GRID_ + xi)) : -1;
        swgt[pt] = feats[pixel * KPAD + d];
    }
    __syncthreads();

    // ---- scatter: lane = channel, 4 points per pass, fp32 L2 atomics ----
    const int c   = tid & 63;
    const int sub = tid >> 6;    // 0..3
    for (int base = 0; base < 16 * DD; base += 4) {
        const int pt = base + sub;
        const int pk = slin[pt];
        if (pk >= 0) {
            const int pixel = pk >> 16;
            const int lin   = pk & 0xFFFF;
            const float fv  = swgt[pt] * feats[pixel * KPAD + DD + c];
            atomicAdd(&out[(size_t)c * (GRID_ * GRID_) + lin], fv);
        }
    }
}

// ---------------------------------------------------------------------------
extern "C" void kernel_launch(void* const* d_in, const int* in_sizes, int n_in,
                              void* d_out, int out_size, void* d_ws, size_t ws_size,
                              hipStream_t stream) {
    const float* x     = (const float*)d_in[0];
    const float* rots  = (const float*)d_in[1];
    const float* trans = (const float*)d_in[2];
    const float* intr  = (const float*)d_in[3];
    const float* Wenc  = (const float*)d_in[4];
    const float* benc  = (const float*)d_in[5];
    float* out = (float*)d_out;
    float* geo = (float*)d_ws;                         // 24*12 floats
    v2f* Wpad  = (v2f*)((float*)d_ws + GEO_FLOATS);    // 256*112 float2 (229KB)

    // out_size = 64*200*200 = 2,560,000 floats -> 2500 blocks x 256 float4s
    lss_zero<<<out_size / 1024, 256, 0, stream>>>((float4*)out);
    lss_prep<<<1, 32, 0, stream>>>(rots, trans, intr, geo);
    lss_pack<<<(CIN / 2) * KPAD / 256, 256, 0, stream>>>(Wenc, Wpad);
    lss_main<<<(B_ * N_) * (HW / 16), 256, 0, stream>>>(x, Wpad, benc, geo, out);
}